// EpisodicMemory_39822936769255
// MI455X (gfx1250) — compile-verified
//
#include <hip/hip_runtime.h>
#include <math.h>

#define NB      512      // stage-1 workgroups
#define B_ROWS  64
#define H_DIM   128
#define S_LEN   32
#define K_TOP   8
#define NEG_INF (-3.0e38f)
#define EPSV    1e-8f

typedef __attribute__((ext_vector_type(2))) float v2f;
typedef __attribute__((ext_vector_type(8))) float v8f;

// compare-exchange keeping the LARGER value at the lower index (descending order)
#define CE(a,b) { if (ts[a] < ts[b]) { float tf_=ts[a]; ts[a]=ts[b]; ts[b]=tf_; \
                                       int   ti_=ti[a]; ti[a]=ti[b]; ti[b]=ti_; } }

// ---------------------------------------------------------------- stage 0
__global__ void em_qnorm(const float* __restrict__ q, float* __restrict__ qinv) {
    int b = threadIdx.x;
    if (b < B_ROWS) {
        float ss = 0.f;
        for (int h = 0; h < H_DIM; ++h) { float v = q[b * H_DIM + h]; ss += v * v; }
        qinv[b] = 1.0f / fmaxf(sqrtf(ss), EPSV);
    }
}

// ---------------------------------------------------------------- stage 1
// 4 waves per block; wave w computes rows [16w,16w+16) of sims for a stream of
// 16-column tiles via V_WMMA_F32_16X16X4_F32. The A operand (query tile) is
// hoisted into 32 v2f registers for the whole kernel; each tile batch-issues
// all 32 B-fragment loads before the WMMA chain so loads pipeline deeply.
__global__ __launch_bounds__(128)
void em_sims_topk(const float* __restrict__ q, const float* __restrict__ ee,
                  const float* __restrict__ qinv, int C,
                  float* __restrict__ pscore, int* __restrict__ pidx)
{
    __shared__ float smem[B_ROWS * 17];
    const int tid  = threadIdx.x;
    const int w    = tid >> 5;
    const int lane = tid & 31;
    const int hi   = lane >> 4;     // half-wave select (K offset +2)
    const int lj   = lane & 15;
    const int wg   = blockIdx.x;
    const int NT   = (C + 15) >> 4;

    // qinv for the 8 accumulator rows this lane holds: M = 16w + v + 8*hi
    float qv[8];
    #pragma unroll
    for (int v = 0; v < 8; ++v) qv[v] = qinv[16 * w + v + 8 * hi];

    // per-row running top-8 (only meaningful for tid < 64)
    float ts[8]; int ti[8];
    #pragma unroll
    for (int j = 0; j < 8; ++j) { ts[j] = NEG_INF; ti[j] = 0; }
    float tmin = NEG_INF; int tminp = 0;

    // A operand: row M = 16w + lj, element K = 4u + 2*hi (+1 via .y).
    // Loop-invariant across the whole column stream -> hoist into registers.
    const float* aBase = q + (size_t)(16 * w + lj) * H_DIM + 2 * hi;
    v2f areg[32];
    #pragma unroll
    for (int u = 0; u < 32; ++u) areg[u] = *(const v2f*)(aBase + 4 * u);

    const int bLane = 2 * hi;            // lane-constant sub-row offset

    for (int t = wg; t < NT; t += NB) {
        const int n0  = t << 4;
        const int col = n0 + lj;
        const int cc  = (col < C) ? col : (C - 1);
        const int bOff = cc * H_DIM + bLane;   // 32-bit offset off uniform base

        // batch-issue all 32 B-fragment loads (stay in flight during WMMA chain)
        v2f breg[32];
        #pragma unroll
        for (int u = 0; u < 32; ++u) breg[u] = *(const v2f*)(ee + bOff + 4 * u);

        // prefetch next tile's B operand (4 x 128B cachelines per 512B row)
        int tn = t + NB;
        if (tn < NT) {
            int coln = (tn << 4) + lj;
            const float* pf = ee + (size_t)((coln < C) ? coln : (C - 1)) * H_DIM;
            __builtin_prefetch(pf +  0, 0, 0);
            __builtin_prefetch(pf + 32, 0, 0);
            __builtin_prefetch(pf + 64, 0, 0);
            __builtin_prefetch(pf + 96, 0, 0);
        }

        v8f   acc = {};
        float ssq = 0.f;
        #pragma unroll
        for (int u = 0; u < 32; ++u) {
            ssq += breg[u].x * breg[u].x + breg[u].y * breg[u].y;
            acc = __builtin_amdgcn_wmma_f32_16x16x4_f32(false, areg[u], false, breg[u],
                                                        (short)0, acc, false, false);
        }

        // column sum-of-squares: fold the two half-waves together
        ssq += __shfl_xor(ssq, 16, 32);
        float einv = 1.0f / fmaxf(sqrtf(ssq), EPSV);

        const bool valid = (col < C);
        #pragma unroll
        for (int v = 0; v < 8; ++v) {
            float s = valid ? (acc[v] * qv[v] * einv) : NEG_INF;
            smem[(16 * w + v + 8 * hi) * 17 + lj] = s;
        }
        __syncthreads();

        if (tid < B_ROWS) {
            #pragma unroll
            for (int j = 0; j < 16; ++j) {
                float v = smem[tid * 17 + j];
                if (v > tmin) {
                    int c2 = n0 + j;
                    #pragma unroll
                    for (int s = 0; s < 8; ++s)
                        if (s == tminp) { ts[s] = v; ti[s] = c2; }
                    tmin = ts[0]; tminp = 0;
                    #pragma unroll
                    for (int s = 1; s < 8; ++s)
                        if (ts[s] < tmin) { tmin = ts[s]; tminp = s; }
                }
            }
        }
        __syncthreads();
    }

    if (tid < B_ROWS) {
        size_t base = ((size_t)wg * B_ROWS + tid) * K_TOP;
        #pragma unroll
        for (int j = 0; j < 8; ++j) { pscore[base + j] = ts[j]; pidx[base + j] = ti[j]; }
    }
}

// ---------------------------------------------------------------- stage 2
// One block per row: merge NB*8 candidates into the final sorted top-8.
__global__ __launch_bounds__(128)
void em_merge(const float* __restrict__ pscore, const int* __restrict__ pidx,
              float* __restrict__ fscore, int* __restrict__ fidx,
              float* __restrict__ out_scores)
{
    __shared__ float ls[128 * 8];
    __shared__ int   li[128 * 8];
    const int b = blockIdx.x;
    const int t = threadIdx.x;

    float ts[8]; int ti[8];
    #pragma unroll
    for (int j = 0; j < 8; ++j) { ts[j] = NEG_INF; ti[j] = 0; }
    float tmin = NEG_INF; int tminp = 0;

    for (int m = t; m < NB * K_TOP; m += 128) {
        int wg = m >> 3, slot = m & 7;
        size_t a = ((size_t)wg * B_ROWS + b) * K_TOP + slot;
        float v = pscore[a];
        if (v > tmin) {
            int c2 = pidx[a];
            #pragma unroll
            for (int s = 0; s < 8; ++s)
                if (s == tminp) { ts[s] = v; ti[s] = c2; }
            tmin = ts[0]; tminp = 0;
            #pragma unroll
            for (int s = 1; s < 8; ++s)
                if (ts[s] < tmin) { tmin = ts[s]; tminp = s; }
        }
    }

    // Batcher odd-even merge sort, 8 elements, descending (19 CEs)
    CE(0,1) CE(2,3) CE(4,5) CE(6,7)
    CE(0,2) CE(1,3) CE(4,6) CE(5,7)
    CE(1,2) CE(5,6)
    CE(0,4) CE(1,5) CE(2,6) CE(3,7)
    CE(2,4) CE(3,5)
    CE(1,2) CE(3,4) CE(5,6)

    #pragma unroll
    for (int j = 0; j < 8; ++j) { ls[t * 8 + j] = ts[j]; li[t * 8 + j] = ti[j]; }
    __syncthreads();

    for (int stride = 64; stride >= 1; stride >>= 1) {
        if (t < stride) {
            int p = (t + stride) * 8;
            float ms[8]; int mi[8];
            #pragma unroll
            for (int j = 0; j < 8; ++j) {          // top-8 of two sorted lists
                float ov = ls[p + 7 - j]; int oi = li[p + 7 - j];
                bool take = (ov > ts[j]);
                ms[j] = take ? ov : ts[j];
                mi[j] = take ? oi : ti[j];
            }
            #pragma unroll
            for (int j = 0; j < 8; ++j) { ts[j] = ms[j]; ti[j] = mi[j]; }
            // bitonic merge network (descending)
            CE(0,4) CE(1,5) CE(2,6) CE(3,7)
            CE(0,2) CE(1,3) CE(4,6) CE(5,7)
            CE(0,1) CE(2,3) CE(4,5) CE(6,7)
            #pragma unroll
            for (int j = 0; j < 8; ++j) { ls[t * 8 + j] = ts[j]; li[t * 8 + j] = ti[j]; }
        }
        __syncthreads();
    }

    if (t == 0) {
        #pragma unroll
        for (int j = 0; j < 8; ++j) { fscore[b * K_TOP + j] = ts[j]; fidx[b * K_TOP + j] = ti[j]; }
        if (b == 0) {
            #pragma unroll
            for (int j = 0; j < 8; ++j) out_scores[j] = ts[j];
        }
    }
}

// ---------------------------------------------------------------- stage 3
__global__ __launch_bounds__(256)
void em_gather(const float* __restrict__ episodes, const int* __restrict__ fidx,
               float* __restrict__ out, int C)
{
    const int r = blockIdx.x;            // rank 0..7 of row 0's top-k
    int idx = fidx[r];
    if (idx < 0) idx = 0;
    if (idx >= C) idx = C - 1;
    const float4* src = (const float4*)(episodes + (size_t)idx * (S_LEN * H_DIM));
    float4*       dst = (float4*)(out + (size_t)r * (S_LEN * H_DIM));
    for (int i = threadIdx.x; i < (S_LEN * H_DIM) / 4; i += blockDim.x) dst[i] = src[i];
}

// ---------------------------------------------------------------- launch
extern "C" void kernel_launch(void* const* d_in, const int* in_sizes, int n_in,
                              void* d_out, int out_size, void* d_ws, size_t ws_size,
                              hipStream_t stream)
{
    const float* q  = (const float*)d_in[0];   // [64,128]
    const float* ep = (const float*)d_in[1];   // [C,32,128]
    const float* ee = (const float*)d_in[2];   // [C,128]
    const int C = in_sizes[2] / H_DIM;

    char* ws = (char*)d_ws;
    const size_t partBytes = (size_t)NB * B_ROWS * K_TOP * 4;
    float* qinv   = (float*)ws;                               // 64 f32
    float* pscore = (float*)(ws + 256);                       // NB*64*8 f32
    int*   pidx   = (int*)  (ws + 256 + partBytes);           // NB*64*8 i32
    float* fscore = (float*)(ws + 256 + 2 * partBytes);       // 64*8 f32
    int*   fidx   = (int*)  (ws + 256 + 2 * partBytes + (size_t)B_ROWS * K_TOP * 4);

    float* out        = (float*)d_out;
    float* out_scores = out + (size_t)K_TOP * S_LEN * H_DIM;  // tail: top_scores[0]

    em_qnorm    <<<1,      64,  0, stream>>>(q, qinv);
    em_sims_topk<<<NB,     128, 0, stream>>>(q, ee, qinv, C, pscore, pidx);
    em_merge    <<<B_ROWS, 128, 0, stream>>>(pscore, pidx, fscore, fidx, out_scores);
    em_gather   <<<K_TOP,  256, 0, stream>>>(ep, fidx, out, C);
}